// BatchBlur_SV_test_4105988735743
// MI455X (gfx1250) — compile-verified
//
#include <hip/hip_runtime.h>
#include <cstdint>

// ---------------------------------------------------------------------------
// 19x19 box-sum blur, reflect padding, fused single pass.
//   grid  = (1024/64, 1024/64, 16*3), block = 256 threads (8 waves, wave32)
//   LDS: 82x83 staged input tile (stride 83 -> conflict-free row-strided reads)
//        82x65 horizontal box sums (stride 65 -> conflict-free strided writes)
//   Horizontal pass: 164 strips (row x half), 32-wide running sum each.
//   Vertical pass:   256 strips (col x quarter), 16-tall running sum each.
// Memory-bound target ~17us @ 23.3 TB/s; VALU kept ~6 adds/output.
// Staging uses CDNA5 GLOBAL_LOAD_ASYNC_TO_LDS_B32 (ASYNCcnt) when available.
// ---------------------------------------------------------------------------

#define LWIN 19
#define PAD 9
#define IMG_W 1024
#define IMG_H 1024
#define TX 64
#define TY 64
#define IN_W (TX + LWIN - 1)   // 82
#define IN_H (TY + LWIN - 1)   // 82
#define IN_STRIDE 83           // gcd(83,64)=1 -> conflict-free across rows
#define H_STRIDE 65            // 64+1       -> conflict-free across rows
#define NTHREADS 256

#if defined(__has_builtin)
#if __has_builtin(__builtin_amdgcn_global_load_async_to_lds_b32)
#define USE_ASYNC_LDS 1
#endif
#endif

#ifdef USE_ASYNC_LDS
typedef __attribute__((address_space(1))) int gas_int;  // global (AS1)
typedef __attribute__((address_space(3))) int las_int;  // LDS (AS3)
#endif

__device__ __forceinline__ int reflect_idx(int t) {
  // jnp.pad mode="reflect": -1 -> 1, 1024 -> 1022 (no edge duplication)
  t = (t < 0) ? -t : t;
  return (t >= IMG_W) ? (2 * (IMG_W - 1) - t) : t;
}

__global__ __launch_bounds__(NTHREADS) void box19_fused_kernel(
    const float* __restrict__ in, float* __restrict__ out) {
  __shared__ float s_in[IN_H][IN_STRIDE];
  __shared__ float s_h[IN_H][H_STRIDE];

  const int tid = threadIdx.x;
  const int x0 = blockIdx.x * TX;
  const int y0 = blockIdx.y * TY;
  const size_t plane = (size_t)blockIdx.z * (size_t)(IMG_W * IMG_H);
  const float* __restrict__ pin = in + plane;
  float* __restrict__ pout = out + plane;

  // ---- Stage (TY+18) x (TX+18) tile into LDS with reflect indexing --------
  for (int e = tid; e < IN_H * IN_W; e += NTHREADS) {
    const int j = e / IN_W;
    const int i = e - j * IN_W;
    const int gy = reflect_idx(y0 - PAD + j);
    const int gx = reflect_idx(x0 - PAD + i);
    const float* src = pin + (size_t)gy * IMG_W + gx;
#ifdef USE_ASYNC_LDS
    // CDNA5 async data mover: global -> LDS without touching VGPRs.
    __builtin_amdgcn_global_load_async_to_lds_b32(
        (gas_int*)(uintptr_t)src,
        (las_int*)(uint32_t)(uintptr_t)&s_in[j][i],
        /*offset=*/0, /*cpol=*/0);
#else
    s_in[j][i] = *src;
#endif
  }
#ifdef USE_ASYNC_LDS
#if __has_builtin(__builtin_amdgcn_s_wait_asynccnt)
  __builtin_amdgcn_s_wait_asynccnt(0);
#else
  asm volatile("s_wait_asynccnt 0" ::: "memory");
#endif
#endif
  __syncthreads();

  // ---- Horizontal running-sum pass: H(j,x) = sum_{d<19} s_in[j][x+d] ------
  // 164 strips: row j in [0,82), half h in {0,1} covering 32 columns each.
  {
    const int strip = tid;
    if (strip < IN_H * 2) {
      const int j = strip >> 1;
      const int hx = (strip & 1) * 32;
      float acc = 0.0f;
#pragma unroll
      for (int d = 0; d < LWIN - 1; ++d) acc += s_in[j][hx + d];
      for (int k = 0; k < 32; ++k) {
        acc += s_in[j][hx + k + LWIN - 1];
        s_h[j][hx + k] = acc;
        acc -= s_in[j][hx + k];
      }
    }
  }
  __syncthreads();

  // ---- Vertical running-sum pass + store ----------------------------------
  // 256 strips: column x in [0,64), quarter q in [0,4) covering 16 rows each.
  {
    const int x = tid & (TX - 1);
    const int ybase = (tid >> 6) * (TY / 4);  // 0,16,32,48
    float acc = 0.0f;
#pragma unroll
    for (int d = 0; d < LWIN - 1; ++d) acc += s_h[ybase + d][x];
    float* orow = pout + (size_t)(y0 + ybase) * IMG_W + (x0 + x);
    for (int r = 0; r < TY / 4; ++r) {
      acc += s_h[ybase + r + LWIN - 1][x];
      orow[0] = acc;
      acc -= s_h[ybase + r][x];
      orow += IMG_W;
    }
  }
}

extern "C" void kernel_launch(void* const* d_in, const int* in_sizes, int n_in,
                              void* d_out, int out_size, void* d_ws, size_t ws_size,
                              hipStream_t stream) {
  (void)in_sizes; (void)n_in; (void)d_ws; (void)ws_size; (void)out_size;
  const float* in = (const float*)d_in[0];
  float* out = (float*)d_out;
  dim3 grid(IMG_W / TX, IMG_H / TY, 16 * 3);
  dim3 block(NTHREADS);
  hipLaunchKernelGGL(box19_fused_kernel, grid, block, 0, stream, in, out);
}